// TensorBasis_18588618457097
// MI455X (gfx1250) — compile-verified
//
#include <hip/hip_runtime.h>
#include <cmath>

typedef __attribute__((ext_vector_type(2))) float v2f;
typedef __attribute__((ext_vector_type(8))) float v8f;

#define CUTOFF     5.0f
#define WIDTH      0.5f
#define SQRT3_4PI  0.4886025119029199f
#define PI_F       3.14159265358979323f
#define INV_SQRT2  0.7071067811865476f

// ---------------------------------------------------------------------------
// Zero the spex accumulator (A * 96 floats) in workspace.
// ---------------------------------------------------------------------------
__global__ void zero_ws_kernel(float* __restrict__ ws, int n) {
    int i = blockIdx.x * blockDim.x + threadIdx.x;
    int stride = gridDim.x * blockDim.x;
    for (; i < n; i += stride) ws[i] = 0.0f;
}

// ---------------------------------------------------------------------------
// Edge scatter: one wave (32 lanes) per edge; lane d = (radial n = d>>2,
// pseudo q = d&3).  Each lane does 1 sqrt, 1 sin, 1 cos and 3 atomic adds
// into contiguous 128B segments of spex[center][m][d].  spex (19.2 MB)
// stays resident in the 192 MB L2, so atomics never hit HBM.
// ---------------------------------------------------------------------------
__global__ __launch_bounds__(256) void edge_kernel(
    const float* __restrict__ vecs,      // (E,3)
    const float* __restrict__ Wsp,       // (T,4)
    const int*   __restrict__ centers,   // (E,)
    const int*   __restrict__ neighbors, // (E,)
    const int*   __restrict__ species,   // (A,)
    float*       __restrict__ spex,      // (A,3,32) accumulator
    int E)
{
    const int lane = threadIdx.x & 31;
    const int wave = threadIdx.x >> 5;
    const int e = blockIdx.x * 8 + wave;
    if (e >= E) return;

    const float vx = vecs[e * 3 + 0];
    const float vy = vecs[e * 3 + 1];
    const float vz = vecs[e * 3 + 2];
    const int ci = centers[e];
    const int ni = neighbors[e];
    const int sp = species[ni];

    const int nidx = lane >> 2;   // radial index 0..7
    const int q    = lane & 3;    // pseudo index 0..3
    const float ps = Wsp[sp * 4 + q];

    const float r  = sqrtf(vx * vx + vy * vy + vz * vz);
    const float rr = r + 1e-10f;

    // shifted-cosine cutoff
    float t = (r - (CUTOFF - WIDTH)) * (1.0f / WIDTH);
    t = fminf(fmaxf(t, 0.0f), 1.0f);
    float f = 0.5f * (1.0f + __cosf(PI_F * t));
    f = (r < CUTOFF) ? f : 0.0f;

    // radial basis for this lane's n
    const float Rn = __sinf((PI_F / CUTOFF) * r * (float)(nidx + 1)) * f / rr;
    const float w  = Rn * ps;

    // l=1 real spherical harmonics in (m=-1,0,1) = (y,z,x)/r order
    const float inv = SQRT3_4PI / rr;
    const float Y0 = inv * vy;
    const float Y1 = inv * vz;
    const float Y2 = inv * vx;

    float* base = spex + (size_t)ci * 96 + lane;
    unsafeAtomicAdd(base +  0, Y0 * w);   // global_atomic_add_f32, coalesced
    unsafeAtomicAdd(base + 32, Y1 * w);
    unsafeAtomicAdd(base + 64, Y2 * w);
}

// ---------------------------------------------------------------------------
// Atom stage: per wave, 16 atoms.  X[mu] = spex[a,mu,:] * Emb[species[a],:]
// (16x32), multiplied by Wc (32x3 padded to 32x16) via eight
// V_WMMA_F32_16X16X4_F32 per mu-tile.  B-fragments are built branchlessly
// (clamped index + 0/1 mask -> v_cndmask, no exec-mask branches) and hoisted
// out of the K loop so the Wc loads form a single clause.  D tiles staged
// through LDS, then lanes 0..15 gather their atom's 3x3 v-matrix and do the
// CG (cross-product) combine for the third basis vector.
// ---------------------------------------------------------------------------
__global__ __launch_bounds__(256) void atom_kernel(
    const float* __restrict__ spex,     // (A,3,32)
    const float* __restrict__ Emb,      // (T,32)
    const float* __restrict__ Wc,       // (32,3)
    const int*   __restrict__ species,  // (A,)
    float*       __restrict__ out,      // (A,3,3)
    int A)
{
    __shared__ float lds[8][3 * 256];   // per-wave: 3 tiles of 16x16 f32

    const int lane = threadIdx.x & 31;
    const int wave = threadIdx.x >> 5;
    const int a_base = (blockIdx.x * 8 + wave) * 16;
    const int arow = a_base + (lane & 15);
    const int a_cl = (arow < A) ? arow : (A - 1);   // clamp reads; EXEC stays all-1s

    const int sp   = species[a_cl];
    const int koff = (lane >> 4) << 1;  // lanes 0-15 -> K {0,1}; 16-31 -> K {2,3}
    const int ncol = lane & 15;

    const float* embp = Emb + sp * 32;
    const float* sx   = spex + (size_t)a_cl * 96;

    // ---- hoisted, branchless B fragments (Wc padded to 16 columns) ----
    const float bmask = (ncol < 3) ? 1.0f : 0.0f;
    const int   nc    = (ncol < 3) ? ncol : 0;      // max index 31*3+2 = 95, in range
    v2f bfrag[8];
#pragma unroll
    for (int kc = 0; kc < 8; ++kc) {
        const int k0 = kc * 4 + koff;
        bfrag[kc][0] = Wc[k0 * 3 + nc] * bmask;
        bfrag[kc][1] = Wc[(k0 + 1) * 3 + nc] * bmask;
    }

    v8f acc0 = {}; v8f acc1 = {}; v8f acc2 = {};

#pragma unroll
    for (int kc = 0; kc < 8; ++kc) {
        const int k0 = kc * 4 + koff;
        const int k1 = k0 + 1;
        const v2f b = bfrag[kc];

        const float e0 = embp[k0];
        const float e1 = embp[k1];

        v2f a0v; a0v[0] = sx[ 0 + k0] * e0; a0v[1] = sx[ 0 + k1] * e1;
        acc0 = __builtin_amdgcn_wmma_f32_16x16x4_f32(false, a0v, false, b,
                                                     (short)0, acc0, false, false);
        v2f a1v; a1v[0] = sx[32 + k0] * e0; a1v[1] = sx[32 + k1] * e1;
        acc1 = __builtin_amdgcn_wmma_f32_16x16x4_f32(false, a1v, false, b,
                                                     (short)0, acc1, false, false);
        v2f a2v; a2v[0] = sx[64 + k0] * e0; a2v[1] = sx[64 + k1] * e1;
        acc2 = __builtin_amdgcn_wmma_f32_16x16x4_f32(false, a2v, false, b,
                                                     (short)0, acc2, false, false);
    }

    // D layout: VGPR i, lane L -> (M = i + 8*(L>=16), N = L%16)
    float* wl = &lds[wave][0];
    const int Mbase = (lane >= 16) ? 8 : 0;
#pragma unroll
    for (int i = 0; i < 8; ++i) {
        const int M = Mbase + i;
        wl[0 * 256 + M * 16 + ncol] = acc0[i];
        wl[1 * 256 + M * 16 + ncol] = acc1[i];
        wl[2 * 256 + M * 16 + ncol] = acc2[i];
    }
    __syncthreads();

    if (lane < 16 && arow < A) {
        float v[3][3];   // v[mu][basis]
#pragma unroll
        for (int mu = 0; mu < 3; ++mu) {
            v[mu][0] = wl[mu * 256 + lane * 16 + 0];
            v[mu][1] = wl[mu * 256 + lane * 16 + 1];
            v[mu][2] = wl[mu * 256 + lane * 16 + 2];
        }
        // CG l=1 x l=1 -> l=1: cross product in cartesian order with
        // (m=-1,0,1) = (y,z,x) permutation, scaled by 1/sqrt(2).
        const float Ax = v[2][0], Ay = v[0][0], Az = v[1][0];
        const float Bx = v[2][1], By = v[0][1], Bz = v[1][1];
        const float cx = Ay * Bz - Az * By;
        const float cy = Az * Bx - Ax * Bz;
        const float cz = Ax * By - Ay * Bx;
        const float v3m[3] = { cy * INV_SQRT2, cz * INV_SQRT2, cx * INV_SQRT2 };

        float* o = out + (size_t)arow * 9;
#pragma unroll
        for (int mu = 0; mu < 3; ++mu) {
            o[mu * 3 + 0] = v[mu][0];
            o[mu * 3 + 1] = v[mu][1];
            o[mu * 3 + 2] = v3m[mu];
        }
    }
}

// ---------------------------------------------------------------------------
extern "C" void kernel_launch(void* const* d_in, const int* in_sizes, int n_in,
                              void* d_out, int out_size, void* d_ws, size_t ws_size,
                              hipStream_t stream) {
    const float* vecs      = (const float*)d_in[0];  // (E,3)
    const float* W_species = (const float*)d_in[1];  // (T,4)
    const float* Emb       = (const float*)d_in[2];  // (T,32)
    const float* Wc        = (const float*)d_in[3];  // (32,3)
    const int*   centers   = (const int*)d_in[4];    // (E,)
    const int*   neighbors = (const int*)d_in[5];    // (E,)
    const int*   species   = (const int*)d_in[6];    // (A,)
    float*       out       = (float*)d_out;          // (A,3,3)

    const int E = in_sizes[0] / 3;
    const int A = in_sizes[6];

    float* spex = (float*)d_ws;                      // A*96 floats (19.2 MB)
    const int n_spex = A * 96;

    {
        int blocks = (n_spex + 255) / 256;
        if (blocks > 16384) blocks = 16384;
        zero_ws_kernel<<<blocks, 256, 0, stream>>>(spex, n_spex);
    }
    {
        int blocks = (E + 7) / 8;                    // 8 waves (edges) per block
        edge_kernel<<<blocks, 256, 0, stream>>>(vecs, W_species, centers,
                                                neighbors, species, spex, E);
    }
    {
        int blocks = (A + 127) / 128;                // 8 waves x 16 atoms per block
        atom_kernel<<<blocks, 256, 0, stream>>>(spex, Emb, Wc, species, out, A);
    }
}